// GCN_37391985279003
// MI455X (gfx1250) — compile-verified
//
#include <hip/hip_runtime.h>
#include <hip/hip_bf16.h>

typedef __attribute__((ext_vector_type(2))) float v2f;
typedef __attribute__((ext_vector_type(8))) float v8f;

#define F_IN 128
#define F_HID 16
#define F_OUT 32

// ---------------- degree / normalization ----------------

__global__ void deg_init_kernel(float* __restrict__ deg, int N) {
    int i = blockIdx.x * blockDim.x + threadIdx.x;
    if (i < N) deg[i] = 1.0f;  // self-loop
}

__global__ void deg_count_kernel(const long long* __restrict__ dst,
                                 float* __restrict__ deg, int E) {
    int e = blockIdx.x * blockDim.x + threadIdx.x;
    if (e < E) unsafeAtomicAdd(&deg[(int)dst[e]], 1.0f);
}

__global__ void deg_rsqrt_kernel(float* __restrict__ deg, int N) {
    int i = blockIdx.x * blockDim.x + threadIdx.x;
    if (i < N) deg[i] = rsqrtf(deg[i]);  // deg >= 1 always
}

// ---------------- layer 1 GEMM: g = dinv .* (x @ W1), s = g (self-loop seed) ----------------
// One wave per 16-row tile, N-tile = full F_HID = 16. K = 128 -> 32 x V_WMMA_F32_16X16X4_F32.

__global__ void gemm1_wmma(const float* __restrict__ x, const float* __restrict__ W1,
                           const float* __restrict__ dinv,
                           float* __restrict__ g, float* __restrict__ s, int N) {
    const int lane = threadIdx.x & 31;
    const int wave = blockIdx.x * (blockDim.x >> 5) + (threadIdx.x >> 5);
    const int row0 = wave * 16;
    if (row0 >= N) return;                 // wave-uniform: EXEC all-1s for WMMA below
    const int m    = lane & 15;
    const int koff = (lane >> 4) << 1;     // lanes 0-15: K+0/1, lanes 16-31: K+2/3
    int row = row0 + m;
    if (row >= N) row = N - 1;             // clamp loads for (non-existent here) tail tile
    const float* xr = x + (size_t)row * F_IN;

    v8f c = {};
#pragma unroll
    for (int k0 = 0; k0 < F_IN; k0 += 4) {
        v2f a = *reinterpret_cast<const v2f*>(xr + k0 + koff);  // A[m][k..k+1]
        v2f b;
        b[0] = W1[(k0 + koff) * F_HID + m];                     // B[k][n]
        b[1] = W1[(k0 + koff + 1) * F_HID + m];
        c = __builtin_amdgcn_wmma_f32_16x16x4_f32(false, a, false, b,
                                                  (short)0, c, false, false);
    }

    const int mb = (lane >> 4) << 3;       // C/D: VGPR v -> row mb+v, col m
#pragma unroll
    for (int v = 0; v < 8; ++v) {
        int r = row0 + mb + v;
        if (r < N) {
            float val = c[v] * dinv[r];
            size_t o = (size_t)r * F_HID + m;
            g[o] = val;
            s[o] = val;
        }
    }
}

// ---------------- layer 2 GEMM: g = dinv .* (h @ W2), s(=d_out) = g ----------------
// K = 16 -> 4 WMMA steps, two N-tiles (cols 0-15, 16-31).

__global__ void gemm2_wmma(const float* __restrict__ h, const float* __restrict__ W2,
                           const float* __restrict__ dinv,
                           float* __restrict__ g, float* __restrict__ s, int N) {
    const int lane = threadIdx.x & 31;
    const int wave = blockIdx.x * (blockDim.x >> 5) + (threadIdx.x >> 5);
    const int row0 = wave * 16;
    if (row0 >= N) return;
    const int m    = lane & 15;
    const int koff = (lane >> 4) << 1;
    int row = row0 + m;
    if (row >= N) row = N - 1;
    const float* hr = h + (size_t)row * F_HID;

    v8f c0 = {}, c1 = {};
#pragma unroll
    for (int k0 = 0; k0 < F_HID; k0 += 4) {
        const int k = k0 + koff;
        v2f a = *reinterpret_cast<const v2f*>(hr + k);
        v2f b0, b1;
        b0[0] = W2[k * F_OUT + m];
        b0[1] = W2[(k + 1) * F_OUT + m];
        b1[0] = W2[k * F_OUT + 16 + m];
        b1[1] = W2[(k + 1) * F_OUT + 16 + m];
        c0 = __builtin_amdgcn_wmma_f32_16x16x4_f32(false, a, false, b0,
                                                   (short)0, c0, false, false);
        c1 = __builtin_amdgcn_wmma_f32_16x16x4_f32(false, a, false, b1,
                                                   (short)0, c1, false, false);
    }

    const int mb = (lane >> 4) << 3;
#pragma unroll
    for (int v = 0; v < 8; ++v) {
        int r = row0 + mb + v;
        if (r < N) {
            float d = dinv[r];
            size_t o = (size_t)r * F_OUT + m;
            float v0 = c0[v] * d, v1 = c1[v] * d;
            g[o] = v0;       g[o + 16] = v1;
            s[o] = v0;       s[o + 16] = v1;
        }
    }
}

// ---------------- edge scatter-add: s[dst] += g[src], float4 granularity ----------------

template <int FEATS>
__global__ void scatter_add_kernel(const long long* __restrict__ srcIdx,
                                   const long long* __restrict__ dstIdx,
                                   const float* __restrict__ g,
                                   float* __restrict__ s, int E) {
    constexpr int FG    = FEATS / 4;               // float4 groups per edge
    constexpr int SHIFT = (FEATS == 16) ? 2 : 3;
    long long t = (long long)blockIdx.x * blockDim.x + threadIdx.x;
    if (t >= (long long)E * FG) return;
    int e = (int)(t >> SHIFT);
    int q = (int)(t & (FG - 1));
    int r = (int)srcIdx[e];
    int c = (int)dstIdx[e];
    float4 val = *reinterpret_cast<const float4*>(g + (size_t)r * FEATS + q * 4);
    float* sp = s + (size_t)c * FEATS + q * 4;
    unsafeAtomicAdd(sp + 0, val.x);
    unsafeAtomicAdd(sp + 1, val.y);
    unsafeAtomicAdd(sp + 2, val.z);
    unsafeAtomicAdd(sp + 3, val.w);
}

// ---------------- layer-1 finish: hrelu = relu(dinv[i]*s1 + b1) ----------------

__global__ void finish1_kernel(const float* __restrict__ s1, const float* __restrict__ dinv,
                               const float* __restrict__ b1, float* __restrict__ hrelu,
                               int N) {
    long long idx = (long long)blockIdx.x * blockDim.x + threadIdx.x;
    if (idx >= (long long)N * F_HID) return;
    int i = (int)(idx >> 4);
    int f = (int)(idx & 15);
    float v = dinv[i] * s1[idx] + b1[f];
    hrelu[idx] = v > 0.0f ? v : 0.0f;
}

// ---------------- layer-2 finish + log_softmax: one wave32 per node, lane == feature ----------------

__global__ void finish2_logsoftmax(const float* __restrict__ dinv,
                                   const float* __restrict__ b2,
                                   float* __restrict__ out, int N) {
    int node = blockIdx.x * (blockDim.x >> 5) + (threadIdx.x >> 5);
    int f = threadIdx.x & 31;
    if (node >= N) return;
    size_t o = (size_t)node * F_OUT + f;
    float v = dinv[node] * out[o] + b2[f];
    float mx = v;
#pragma unroll
    for (int off = 16; off > 0; off >>= 1) mx = fmaxf(mx, __shfl_xor(mx, off, 32));
    float ex = __expf(v - mx);
    float sum = ex;
#pragma unroll
    for (int off = 16; off > 0; off >>= 1) sum += __shfl_xor(sum, off, 32);
    out[o] = v - mx - __logf(sum);
}

// ---------------- launcher ----------------

extern "C" void kernel_launch(void* const* d_in, const int* in_sizes, int n_in,
                              void* d_out, int out_size, void* d_ws, size_t ws_size,
                              hipStream_t stream) {
    (void)n_in; (void)out_size; (void)ws_size;
    const float*     x   = (const float*)d_in[0];
    const long long* ei  = (const long long*)d_in[1];   // int64 [2, E]
    const float*     W1  = (const float*)d_in[2];
    const float*     b1  = (const float*)d_in[3];
    const float*     W2  = (const float*)d_in[4];
    const float*     b2  = (const float*)d_in[5];
    float*           out = (float*)d_out;

    const int N = in_sizes[0] / F_IN;
    const int E = in_sizes[1] / 2;
    const long long* src = ei;        // edge_index[0] : sources
    const long long* dst = ei + E;    // edge_index[1] : targets

    // workspace layout (floats): dinv[N] | g1/hrelu[16N] | s1[16N] ... g2[32N] overlays dead s1
    float* ws   = (float*)d_ws;
    float* dinv = ws;
    float* g1   = ws + (size_t)N;          // 16N; reused as hrelu after finish1
    float* s1   = ws + (size_t)17 * N;     // 16N
    float* g2   = ws + (size_t)17 * N;     // 32N, overwrites dead s1 region (gemm2 runs after finish1)

    const int T = 256;  // 8 wave32s per block
    // degrees -> dinv
    deg_init_kernel <<<(N + T - 1) / T, T, 0, stream>>>(dinv, N);
    deg_count_kernel<<<(E + T - 1) / T, T, 0, stream>>>(dst, dinv, E);
    deg_rsqrt_kernel<<<(N + T - 1) / T, T, 0, stream>>>(dinv, N);

    const int tiles  = (N + 15) >> 4;
    const int gblk   = (tiles + 7) / 8;     // 8 waves (tiles) per block

    // layer 1
    gemm1_wmma<<<gblk, T, 0, stream>>>(x, W1, dinv, g1, s1, N);
    long long t1 = (long long)E * 4;
    scatter_add_kernel<F_HID><<<(unsigned)((t1 + T - 1) / T), T, 0, stream>>>(src, dst, g1, s1, E);
    long long n16 = (long long)N * F_HID;
    finish1_kernel<<<(unsigned)((n16 + T - 1) / T), T, 0, stream>>>(s1, dinv, b1, g1, N);

    // layer 2 (accumulate directly into d_out)
    gemm2_wmma<<<gblk, T, 0, stream>>>(g1, W2, dinv, g2, out, N);
    long long t2 = (long long)E * 8;
    scatter_add_kernel<F_OUT><<<(unsigned)((t2 + T - 1) / T), T, 0, stream>>>(src, dst, g2, out, E);
    finish2_logsoftmax<<<(N + 7) / 8, T, 0, stream>>>(dinv, b2, out, N);
}